// BN_78735340470499
// MI455X (gfx1250) — compile-verified
//
#include <hip/hip_runtime.h>

// ---------------------------------------------------------------------------
// Reference: u = colsum(x*x) + eps ; out = x * rsqrt(u).  x: [32768, 2048] f32.
// Memory-bound: 768 MB min traffic -> ~33 us @ 23.3 TB/s.  Compute negligible.
// Strategy:
//   K1: column sum-of-squares via V_WMMA_F32_16X16X4_F32 with A = ones(16x4):
//       D = ones x B + C  => every row of D is the column sums of B(4x16).
//       Exact f32 math; reduction runs on the matrix pipe. Deterministic
//       per-row-chunk partials (no float atomics).
//   K2: ordered reduce of 64 chunk partials, + eps, rsqrt -> inv[2048].
//   K3: streaming float4 scale: out = x * inv[col].
// ---------------------------------------------------------------------------

typedef __attribute__((ext_vector_type(2))) float v2f;
typedef __attribute__((ext_vector_type(4))) float f4;
typedef __attribute__((ext_vector_type(8))) float v8f;

#define NROWS 32768
#define NCOLS 2048
#define ROWS_PER_CHUNK 512
#define NUM_CHUNKS (NROWS / ROWS_PER_CHUNK)   // 64
#define EPS 1e-6f

// ---- Kernel 1: per-chunk column sum-of-squares on the WMMA pipe -----------
// Block = 256 threads = 8 waves; each wave owns 32 columns (2 WMMA tiles).
// Grid = (NCOLS/256, NUM_CHUNKS).  Per 4-row step the wave's four 32-lane
// loads cover full 128B cachelines of rows r..r+3 (cols cbase..cbase+31).
__global__ __launch_bounds__(256) void colsumsq_wmma(const float* __restrict__ x,
                                                     float* __restrict__ partial) {
  const int lane = threadIdx.x & 31;
  const int wave = threadIdx.x >> 5;
  const int n    = lane & 15;   // column within 16-wide tile
  const int half = lane >> 4;   // lanes 0-15 feed rows r,r+1 ; lanes 16-31 rows r+2,r+3
  const int cbase = blockIdx.x * 256 + wave * 32;
  const int col0 = cbase + n;        // tile-0 column for this lane
  const int col1 = cbase + 16 + n;   // tile-1 column for this lane

  const float* px = x + (long)blockIdx.y * ROWS_PER_CHUNK * NCOLS;

  const v2f ones = {1.0f, 1.0f};     // A matrix: all ones (layout-invariant)
  v8f acc0 = {};
  v8f acc1 = {};

  for (int r = 0; r < ROWS_PER_CHUNK; r += 4) {
    const long ro0 = (long)(r + half * 2) * NCOLS;  // first row this lane-half feeds
    const long ro1 = ro0 + NCOLS;                   // second row

    // prefetch next 4-row group (speculative; harmless at the tail)
    __builtin_prefetch(&px[ro0 + col0 + 4 * NCOLS], 0, 3);

    float a0 = px[ro0 + col0];
    float a1 = px[ro1 + col0];
    float b0 = px[ro0 + col1];
    float b1 = px[ro1 + col1];

    v2f B0 = { a0 * a0, a1 * a1 };   // squared elements -> B(4x16) tile 0
    v2f B1 = { b0 * b0, b1 * b1 };   // tile 1

    // D = ones(16x4) x B(4x16) + C : every D row = column sums of B.
    acc0 = __builtin_amdgcn_wmma_f32_16x16x4_f32(false, ones, false, B0,
                                                 (short)0, acc0, false, false);
    acc1 = __builtin_amdgcn_wmma_f32_16x16x4_f32(false, ones, false, B1,
                                                 (short)0, acc1, false, false);
  }

  // acc VGPR0, lanes 0-15 hold D[M=0][N=lane] = column sum for col cbase+lane.
  if (lane < 16) {
    float* p = partial + (long)blockIdx.y * NCOLS;
    p[col0] = acc0[0];
    p[col1] = acc1[0];
  }
}

// ---- Kernel 2: deterministic chunk reduce + rsqrt -------------------------
__global__ __launch_bounds__(256) void reduce_rsqrt(const float* __restrict__ partial,
                                                    float* __restrict__ inv) {
  const int j = blockIdx.x * blockDim.x + threadIdx.x;
  if (j >= NCOLS) return;
  float s = 0.0f;
  for (int c = 0; c < NUM_CHUNKS; ++c)
    s += partial[(long)c * NCOLS + j];
  inv[j] = rsqrtf(s + EPS);
}

// ---- Kernel 3: streaming float4 scale -------------------------------------
__global__ __launch_bounds__(256) void scale_cols(const float* __restrict__ x,
                                                  const float* __restrict__ inv,
                                                  float* __restrict__ out) {
  const long t = (long)blockIdx.x * blockDim.x + threadIdx.x;  // float4 index
  const f4* __restrict__ x4 = (const f4*)x;
  const f4* __restrict__ i4 = (const f4*)inv;
  f4* __restrict__ o4 = (f4*)out;
  const int col4 = (int)(t & (NCOLS / 4 - 1));  // column/4 within row
  o4[t] = x4[t] * i4[col4];
}

extern "C" void kernel_launch(void* const* d_in, const int* in_sizes, int n_in,
                              void* d_out, int out_size, void* d_ws, size_t ws_size,
                              hipStream_t stream) {
  const float* x = (const float*)d_in[0];
  float* out = (float*)d_out;

  // Workspace layout: [NUM_CHUNKS * NCOLS] partials, then [NCOLS] inv table.
  float* partial = (float*)d_ws;
  float* inv = partial + (size_t)NUM_CHUNKS * NCOLS;

  dim3 g1(NCOLS / 256, NUM_CHUNKS);            // 8 x 64 = 512 blocks
  colsumsq_wmma<<<g1, 256, 0, stream>>>(x, partial);

  reduce_rsqrt<<<(NCOLS + 255) / 256, 256, 0, stream>>>(partial, inv);

  const long total4 = (long)NROWS * NCOLS / 4; // 16M float4s
  scale_cols<<<(int)(total4 / 256), 256, 0, stream>>>(x, inv, out);
}